// Rank_CLS_Loss_22316650070598
// MI455X (gfx1250) — compile-verified
//
#include <hip/hip_runtime.h>
#include <math.h>

#define BSZ     256
#define WAVE    32
#define NWAVES  (BSZ / WAVE)
#define CHUNKS  64
#define TPW     128          // u32 elements per async tile per wave (32 lanes x 16B)
#define L0BITS  11
#define L1BITS  11
#define L2BITS  10
#define L0BINS  2048
#define L1BINS  2048
#define L2BINS  1024
#define CAPK    16384        // per-sample compaction capacity (fits in LDS)

#define LN_HALF (-0.69314718055994531f)

struct SState {
    unsigned cnt_pos, cnt_neg, cnt_hard, max_neg_key;
    float    sum_pos;
    unsigned k, flag_k0, prefix;     // prefix grows: 11b (p0) -> 22b (p01)
    unsigned rem, thresh_key, rem_eq, cnt_app;
    float    sel_num, sel_den;
    unsigned compact, pad2;
};  // 64 bytes

// ---------- float ordering keys ----------
__device__ __forceinline__ unsigned f2key(float f) {
    unsigned b = __float_as_uint(f);
    return (b & 0x80000000u) ? ~b : (b | 0x80000000u);
}
__device__ __forceinline__ float key2f(unsigned k) {
    unsigned b = (k & 0x80000000u) ? (k ^ 0x80000000u) : ~k;
    return __uint_as_float(b);
}
__device__ __forceinline__ unsigned load_key_scalar(int pack, const unsigned* __restrict__ keys,
                                                    const float2* __restrict__ inp2,
                                                    const int* __restrict__ lab,
                                                    size_t base, int i) {
    if (pack) return keys[base + (size_t)i];
    int l = lab[base + (size_t)i];
    if (l != 0) return 0u;
    return f2key(inp2[base + (size_t)i].y);
}

// ---------- CDNA5 async global->LDS (per-lane 16B), tracked by ASYNCcnt ----------
__device__ __forceinline__ void async_ld_b128(unsigned lds_byte_addr, unsigned long long gaddr) {
    asm volatile("global_load_async_to_lds_b128 %0, %1, off"
                 :: "v"(lds_byte_addr), "v"(gaddr)
                 : "memory");
}
__device__ __forceinline__ void wait_async_le1() {
    asm volatile("s_wait_asynccnt 0x1" ::: "memory");
}
__device__ __forceinline__ void wait_async_0() {
    asm volatile("s_wait_asynccnt 0x0" ::: "memory");
}
__device__ __forceinline__ unsigned lds_addr_of(const void* p) {
    return (unsigned)(size_t)p;          // generic LDS pointer: low 32 bits = LDS offset
}

// Stream packed keys through LDS with double-buffered async tiles; call f(key) per element.
// Requires (iend-ibeg) % (NWAVES*TPW) == 0.
template <typename F>
__device__ __forceinline__ void stream_keys_async(const unsigned* __restrict__ keys, size_t base,
                                                  int ibeg, int iend, unsigned* tiles, F f) {
    const int lane = threadIdx.x & (WAVE - 1);
    const int wid  = threadIdx.x / WAVE;
    const int per  = (iend - ibeg) / NWAVES;
    const int ntiles = per / TPW;
    unsigned* buf = tiles + wid * 2 * TPW;                 // this wave's 2 x 512B buffers
    const unsigned lds0 = lds_addr_of(buf);
    const unsigned* g0 = keys + base + (size_t)ibeg + (size_t)wid * per;

    async_ld_b128(lds0 + lane * 16u,
                  (unsigned long long)(size_t)(g0 + lane * 4));
    for (int t = 0; t < ntiles; ++t) {
        if (t + 1 < ntiles) {
            async_ld_b128(lds0 + ((unsigned)(t + 1) & 1u) * (TPW * 4u) + lane * 16u,
                          (unsigned long long)(size_t)(g0 + (size_t)(t + 1) * TPW + lane * 4));
            wait_async_le1();            // tile t complete
        } else {
            wait_async_0();
        }
        const uint4 k4 = *(const uint4*)&buf[(t & 1) * TPW + lane * 4];
        f(k4.x); f(k4.y); f(k4.z); f(k4.w);
    }
}

// ---------- kernels ----------
__global__ void k_init(unsigned* __restrict__ ws, size_t words) {
    size_t i = (size_t)blockIdx.x * blockDim.x + threadIdx.x;
    size_t stride = (size_t)gridDim.x * blockDim.x;
    for (; i < words; i += stride) ws[i] = 0u;
}

__global__ void k_pass0(const float2* __restrict__ inp2, const int* __restrict__ lab,
                        unsigned* __restrict__ keys, SState* __restrict__ st,
                        unsigned* __restrict__ hist0, int N, int pack) {
    __shared__ unsigned h[L0BINS];
    __shared__ unsigned s_pos, s_neg, s_hard, s_maxk;
    __shared__ float s_sum;
    const int s = blockIdx.x / CHUNKS;
    const int c = blockIdx.x % CHUNKS;
    const int chunk = (N + CHUNKS - 1) / CHUNKS;
    const int ibeg = c * chunk;
    const int iend = min(ibeg + chunk, N);
    const size_t base = (size_t)s * N;

    for (int i = threadIdx.x; i < L0BINS; i += BSZ) h[i] = 0u;
    if (threadIdx.x == 0) { s_pos = 0u; s_neg = 0u; s_hard = 0u; s_maxk = 0u; s_sum = 0.f; }
    __syncthreads();

    unsigned lpos = 0, lneg = 0, lhard = 0, lmax = 0;
    float lsum = 0.f;

    if ((chunk % (4 * BSZ)) == 0 && iend == ibeg + chunk) {
        for (int i = ibeg + threadIdx.x * 4; i < iend; i += 4 * BSZ) {
            const int4 l4 = *(const int4*)&lab[base + (size_t)i];
            const float4* p4 = (const float4*)(inp2 + base + (size_t)i);
            const float4 a = p4[0], b = p4[1];
            const float pr[4] = { a.y, a.w, b.y, b.w };
            const int   ll[4] = { l4.x, l4.y, l4.z, l4.w };
            unsigned ko[4];
#pragma unroll
            for (int j = 0; j < 4; ++j) {
                const unsigned key = f2key(pr[j]);
                unsigned kout = 0u;
                if (ll[j] == 1) { lpos++; lsum += __expf(pr[j]); }
                else {
                    lneg++;
                    if (pr[j] > LN_HALF) lhard++;
                    lmax = max(lmax, key);
                    atomicAdd(&h[key >> (32 - L0BITS)], 1u);
                    kout = key;
                }
                ko[j] = kout;
            }
            if (pack) *(uint4*)&keys[base + (size_t)i] = make_uint4(ko[0], ko[1], ko[2], ko[3]);
        }
    } else {
        for (int i = ibeg + threadIdx.x; i < iend; i += BSZ) {
            int   l    = lab[base + (size_t)i];
            float pred = inp2[base + (size_t)i].y;
            unsigned key  = f2key(pred);
            unsigned kout = 0u;
            if (l == 1) { lpos++; lsum += __expf(pred); }
            else {
                lneg++;
                if (pred > LN_HALF) lhard++;
                lmax = max(lmax, key);
                atomicAdd(&h[key >> (32 - L0BITS)], 1u);
                kout = key;
            }
            if (pack) keys[base + (size_t)i] = kout;
        }
    }

    atomicAdd(&s_pos, lpos);
    atomicAdd(&s_neg, lneg);
    atomicAdd(&s_hard, lhard);
    atomicMax(&s_maxk, lmax);
    atomicAdd(&s_sum, lsum);
    __syncthreads();

    unsigned* gh = hist0 + (size_t)s * L0BINS;
    for (int i = threadIdx.x; i < L0BINS; i += BSZ)
        if (h[i]) atomicAdd(&gh[i], h[i]);
    if (threadIdx.x == 0) {
        atomicAdd(&st[s].cnt_pos,  s_pos);
        atomicAdd(&st[s].cnt_neg,  s_neg);
        atomicAdd(&st[s].cnt_hard, s_hard);
        atomicMax(&st[s].max_neg_key, s_maxk);
        atomicAdd(&st[s].sum_pos,  s_sum);
    }
}

template <int NB>
__device__ __forceinline__ void walk_bins(const unsigned* __restrict__ ghist, int s,
                                          unsigned rem_in, unsigned* out_bin, unsigned* out_rem) {
    __shared__ unsigned bins[NB];
    __shared__ unsigned grp[BSZ];
    __shared__ unsigned r_bin, r_rem;
    const int GP = NB / BSZ;
    for (int i = threadIdx.x; i < NB; i += BSZ) bins[i] = ghist[(size_t)s * NB + i];
    __syncthreads();
    unsigned sum = 0;
    for (int j = 0; j < GP; j++) sum += bins[threadIdx.x * GP + j];
    grp[threadIdx.x] = sum;
    __syncthreads();
    if (threadIdx.x == 0) {
        unsigned rem = rem_in;
        int gi = BSZ - 1;
        while (gi > 0 && rem > grp[gi]) { rem -= grp[gi]; gi--; }
        int b = gi * GP + GP - 1;
        while (b > gi * GP && rem > bins[b]) { rem -= bins[b]; b--; }
        r_bin = (unsigned)b;
        r_rem = rem;
    }
    __syncthreads();
    *out_bin = r_bin;
    *out_rem = r_rem;
}

__global__ void k_select0(SState* __restrict__ st, const unsigned* __restrict__ hist0) {
    const int s = blockIdx.x;
    __shared__ unsigned sk;
    if (threadIdx.x == 0) {
        unsigned np = st[s].cnt_pos, nn = st[s].cnt_neg, nh = st[s].cnt_hard;
        unsigned k = (np > 0u) ? min(np, nn) : min(max(nh, 8u), nn);
        st[s].k = k;
        sk = k;
        if (k == 0u) { st[s].flag_k0 = 1u; st[s].thresh_key = 0xFFFFFFFFu; st[s].rem_eq = 0u; }
        else st[s].flag_k0 = 0u;
    }
    __syncthreads();
    if (sk == 0u) return;
    unsigned b, rem;
    walk_bins<L0BINS>(hist0, s, sk, &b, &rem);
    if (threadIdx.x == 0) { st[s].prefix = b; st[s].rem = rem; }
}

// pass1: keys strictly above bucket p0 -> accumulate softmax terms directly;
//        keys inside bucket p0 -> mid-11-bit histogram (fallback) + compaction append.
__global__ void k_pass1(const float2* __restrict__ inp2, const int* __restrict__ lab,
                        const unsigned* __restrict__ keys, SState* __restrict__ st,
                        unsigned* __restrict__ hist1, unsigned* __restrict__ cbuf,
                        int N, int pack, int use_cbuf) {
    __shared__ unsigned h[L1BINS];
    __shared__ __align__(16) unsigned tiles[NWAVES * 2 * TPW];
    __shared__ float s_num, s_den;
    const int s = blockIdx.x / CHUNKS;
    if (st[s].flag_k0) return;
    const unsigned p0 = st[s].prefix;
    const float m_prob = __expf(key2f(st[s].max_neg_key));
    const int c = blockIdx.x % CHUNKS;
    const int chunk = (N + CHUNKS - 1) / CHUNKS;
    const int ibeg = c * chunk;
    const int iend = min(ibeg + chunk, N);
    const size_t base = (size_t)s * N;
    for (int i = threadIdx.x; i < L1BINS; i += BSZ) h[i] = 0u;
    if (threadIdx.x == 0) { s_num = 0.f; s_den = 0.f; }
    __syncthreads();

    float lnum = 0.f, lden = 0.f;
    auto body = [&](unsigned key) {
        const unsigned top = key >> (32 - L0BITS);
        if (top > p0) {
            float v = __expf(key2f(key));
            float e = __expf(v - m_prob);
            lnum += v * e;
            lden += e;
        } else if (top == p0) {
            atomicAdd(&h[(key >> L2BITS) & (L1BINS - 1)], 1u);
            if (use_cbuf) {
                unsigned idx = atomicAdd(&st[s].cnt_app, 1u);
                if (idx < CAPK) cbuf[(size_t)s * CAPK + idx] = key;
            }
        }
    };
    if (pack && ((iend - ibeg) % (NWAVES * TPW)) == 0) {
        stream_keys_async(keys, base, ibeg, iend, tiles, body);
    } else {
        for (int i = ibeg + threadIdx.x; i < iend; i += BSZ)
            body(load_key_scalar(pack, keys, inp2, lab, base, i));
    }
    atomicAdd(&s_num, lnum);
    atomicAdd(&s_den, lden);
    __syncthreads();

    unsigned* gh = hist1 + (size_t)s * L1BINS;
    for (int i = threadIdx.x; i < L1BINS; i += BSZ)
        if (h[i]) atomicAdd(&gh[i], h[i]);
    if (threadIdx.x == 0) {
        atomicAdd(&st[s].sel_num, s_num);
        atomicAdd(&st[s].sel_den, s_den);
    }
}

// k_tail: finish the in-bucket selection entirely in LDS from the compacted list.
// Runs the two remaining radix levels in-block, then accumulates in-bucket terms + ties.
__global__ void k_tail(const unsigned* __restrict__ cbuf, SState* __restrict__ st, int use_cbuf) {
    const int s = blockIdx.x;
    if (!use_cbuf) return;
    if (st[s].flag_k0) return;
    const unsigned napp = st[s].cnt_app;
    if (napp > CAPK) return;                     // fallback chain will handle
    __shared__ unsigned kl[CAPK];
    __shared__ unsigned h1[L1BINS];
    __shared__ unsigned h2[L2BINS];
    __shared__ unsigned grp[BSZ];
    __shared__ unsigned sh_bin, sh_rem;
    __shared__ float s_num, s_den;
    const unsigned bc = napp;
    const unsigned p0 = st[s].prefix;            // 11-bit
    const unsigned rem0 = st[s].rem;
    const float m_prob = __expf(key2f(st[s].max_neg_key));
    if (threadIdx.x == 0) { st[s].compact = 1u; s_num = 0.f; s_den = 0.f; }
    for (unsigned i = threadIdx.x; i < bc; i += BSZ) kl[i] = cbuf[(size_t)s * CAPK + i];
    for (int i = threadIdx.x; i < L1BINS; i += BSZ) h1[i] = 0u;
    for (int i = threadIdx.x; i < L2BINS; i += BSZ) h2[i] = 0u;
    __syncthreads();

    // phase A: mid-11 histogram + walk
    for (unsigned i = threadIdx.x; i < bc; i += BSZ)
        atomicAdd(&h1[(kl[i] >> L2BITS) & (L1BINS - 1)], 1u);
    __syncthreads();
    {
        const int GP = L1BINS / BSZ;  // 8
        unsigned sum = 0;
        for (int j = 0; j < GP; j++) sum += h1[threadIdx.x * GP + j];
        grp[threadIdx.x] = sum;
        __syncthreads();
        if (threadIdx.x == 0) {
            unsigned rem = rem0;
            int gi = BSZ - 1;
            while (gi > 0 && rem > grp[gi]) { rem -= grp[gi]; gi--; }
            int b = gi * GP + GP - 1;
            while (b > gi * GP && rem > h1[b]) { rem -= h1[b]; b--; }
            sh_bin = (unsigned)b;
            sh_rem = rem;
        }
        __syncthreads();
    }
    const unsigned p1 = sh_bin;
    const unsigned rem1 = sh_rem;
    __syncthreads();                              // grp reuse below

    // phase B: low-10 histogram within (p0,p1) + walk
    for (unsigned i = threadIdx.x; i < bc; i += BSZ) {
        unsigned key = kl[i];
        if (((key >> L2BITS) & (L1BINS - 1)) == p1)
            atomicAdd(&h2[key & (L2BINS - 1)], 1u);
    }
    __syncthreads();
    {
        const int GP = L2BINS / BSZ;  // 4
        unsigned sum = 0;
        for (int j = 0; j < GP; j++) sum += h2[threadIdx.x * GP + j];
        grp[threadIdx.x] = sum;
        __syncthreads();
        if (threadIdx.x == 0) {
            unsigned rem = rem1;
            int gi = BSZ - 1;
            while (gi > 0 && rem > grp[gi]) { rem -= grp[gi]; gi--; }
            int b = gi * GP + GP - 1;
            while (b > gi * GP && rem > h2[b]) { rem -= h2[b]; b--; }
            sh_bin = (unsigned)b;
            sh_rem = rem;
        }
        __syncthreads();
    }
    const unsigned bt = sh_bin;
    const unsigned rem_eq = sh_rem;
    const unsigned tk = (p0 << (L1BITS + L2BITS)) | (p1 << L2BITS) | bt;

    // phase C: in-bucket strictly-greater terms + ties
    float lnum = 0.f, lden = 0.f;
    for (unsigned i = threadIdx.x; i < bc; i += BSZ) {
        unsigned key = kl[i];
        if (key > tk) {
            float v = __expf(key2f(key));
            float e = __expf(v - m_prob);
            lnum += v * e;
            lden += e;
        }
    }
    atomicAdd(&s_num, lnum);
    atomicAdd(&s_den, lden);
    __syncthreads();
    if (threadIdx.x == 0) {
        float vt = __expf(key2f(tk));
        float et = __expf(vt - m_prob);
        float req = (float)rem_eq;
        st[s].thresh_key = tk;
        st[s].rem_eq = rem_eq;
        st[s].sel_num += s_num + req * vt * et;   // single writer after pass1
        st[s].sel_den += s_den + req * et;
    }
}

__global__ void k_select1(SState* __restrict__ st, const unsigned* __restrict__ hist1) {
    const int s = blockIdx.x;
    if (st[s].flag_k0 || st[s].compact) return;
    unsigned b, rem;
    walk_bins<L1BINS>(hist1, s, st[s].rem, &b, &rem);
    if (threadIdx.x == 0) {
        st[s].prefix = (st[s].prefix << L1BITS) | b;   // 22-bit prefix p01
        st[s].rem = rem;
    }
}

// fallback pass2 (only when compaction overflowed): keys above bucket p01 but inside
// bucket p0 -> accumulate; keys inside p01 -> low-10 histogram (bin == exact key).
__global__ void k_pass2(const float2* __restrict__ inp2, const int* __restrict__ lab,
                        const unsigned* __restrict__ keys, SState* __restrict__ st,
                        unsigned* __restrict__ hist2, int N, int pack) {
    __shared__ unsigned h[L2BINS];
    __shared__ __align__(16) unsigned tiles[NWAVES * 2 * TPW];
    __shared__ float s_num, s_den;
    const int s = blockIdx.x / CHUNKS;
    if (st[s].flag_k0 || st[s].compact) return;
    const unsigned p01 = st[s].prefix;           // 22-bit
    const unsigned p0  = p01 >> L1BITS;          // 11-bit
    const float m_prob = __expf(key2f(st[s].max_neg_key));
    const int c = blockIdx.x % CHUNKS;
    const int chunk = (N + CHUNKS - 1) / CHUNKS;
    const int ibeg = c * chunk;
    const int iend = min(ibeg + chunk, N);
    const size_t base = (size_t)s * N;
    for (int i = threadIdx.x; i < L2BINS; i += BSZ) h[i] = 0u;
    if (threadIdx.x == 0) { s_num = 0.f; s_den = 0.f; }
    __syncthreads();

    float lnum = 0.f, lden = 0.f;
    auto body = [&](unsigned key) {
        const unsigned hi22 = key >> L2BITS;
        if (hi22 == p01) {
            atomicAdd(&h[key & (L2BINS - 1)], 1u);
        } else if (hi22 > p01 && (key >> (32 - L0BITS)) == p0) {
            float v = __expf(key2f(key));
            float e = __expf(v - m_prob);
            lnum += v * e;
            lden += e;
        }
    };
    if (pack && ((iend - ibeg) % (NWAVES * TPW)) == 0) {
        stream_keys_async(keys, base, ibeg, iend, tiles, body);
    } else {
        for (int i = ibeg + threadIdx.x; i < iend; i += BSZ)
            body(load_key_scalar(pack, keys, inp2, lab, base, i));
    }
    atomicAdd(&s_num, lnum);
    atomicAdd(&s_den, lden);
    __syncthreads();

    unsigned* gh = hist2 + (size_t)s * L2BINS;
    for (int i = threadIdx.x; i < L2BINS; i += BSZ)
        if (h[i]) atomicAdd(&gh[i], h[i]);
    if (threadIdx.x == 0) {
        atomicAdd(&st[s].sel_num, s_num);
        atomicAdd(&st[s].sel_den, s_den);
    }
}

// fallback select2: threshold bin + analytic contributions (bin == exact key here).
__global__ void k_select2(SState* __restrict__ st, const unsigned* __restrict__ hist2) {
    const int s = blockIdx.x;
    if (st[s].flag_k0 || st[s].compact) return;
    __shared__ unsigned bins[L2BINS];
    __shared__ unsigned grp[BSZ];
    __shared__ unsigned r_bin, r_rem;
    __shared__ float s_num, s_den;
    const int GP = L2BINS / BSZ;   // 4
    for (int i = threadIdx.x; i < L2BINS; i += BSZ)
        bins[i] = hist2[(size_t)s * L2BINS + i];
    if (threadIdx.x == 0) { s_num = 0.f; s_den = 0.f; }
    __syncthreads();
    unsigned sum = 0;
    for (int j = 0; j < GP; j++) sum += bins[threadIdx.x * GP + j];
    grp[threadIdx.x] = sum;
    __syncthreads();
    if (threadIdx.x == 0) {
        unsigned rem = st[s].rem;
        int gi = BSZ - 1;
        while (gi > 0 && rem > grp[gi]) { rem -= grp[gi]; gi--; }
        int b = gi * GP + GP - 1;
        while (b > gi * GP && rem > bins[b]) { rem -= bins[b]; b--; }
        r_bin = (unsigned)b;
        r_rem = rem;
    }
    __syncthreads();
    const unsigned p01 = st[s].prefix;
    const unsigned bt  = r_bin;
    const float m_prob = __expf(key2f(st[s].max_neg_key));
    float lnum = 0.f, lden = 0.f;
    for (int b = (int)bt + 1 + threadIdx.x; b < L2BINS; b += BSZ) {
        unsigned cb = bins[b];
        if (cb) {
            float v = __expf(key2f((p01 << L2BITS) | (unsigned)b));
            float e = __expf(v - m_prob);
            lnum += (float)cb * v * e;
            lden += (float)cb * e;
        }
    }
    atomicAdd(&s_num, lnum);
    atomicAdd(&s_den, lden);
    __syncthreads();
    if (threadIdx.x == 0) {
        const unsigned tk = (p01 << L2BITS) | bt;
        float vt = __expf(key2f(tk));
        float et = __expf(vt - m_prob);
        float req = (float)r_rem;
        st[s].thresh_key = tk;
        st[s].rem_eq = r_rem;
        st[s].sel_num += s_num + req * vt * et;
        st[s].sel_den += s_den + req * et;
    }
}

__global__ void k_final(const SState* __restrict__ st, float* __restrict__ out, int B) {
    __shared__ float sl[64], sv[64];
    const int s = threadIdx.x;
    float loss = 0.f, val = 0.f;
    if (s < B) {
        SState S = st[s];
        float dist = 0.f;
        if (S.k > 0u)
            dist = S.sel_num / fmaxf(S.sel_den, 1e-30f);
        float x;
        if (S.cnt_pos > 0u) {
            float pos_dist = S.sum_pos / (float)max(S.cnt_pos, 1u);
            x = dist - pos_dist + 0.5f;
        } else {
            x = dist - 1.0f + 0.5f;
        }
        float y = 4.0f * x;                 // stable log1p(exp(y))/4
        loss = (fmaxf(y, 0.f) + log1pf(__expf(-fabsf(y)))) * 0.25f;
        val = (S.cnt_hard > 0u) ? 1.f : 0.f;
        loss *= val;
    }
    sl[threadIdx.x] = loss;
    sv[threadIdx.x] = val;
    __syncthreads();
    if (threadIdx.x == 0) {
        float tl = 0.f, tv = 0.f;
        for (int i = 0; i < 64; i++) { tl += sl[i]; tv += sv[i]; }
        out[0] = (tv > 0.f) ? tl / tv : 0.f;
    }
}

extern "C" void kernel_launch(void* const* d_in, const int* in_sizes, int n_in,
                              void* d_out, int out_size, void* d_ws, size_t ws_size,
                              hipStream_t stream) {
    const int B = 64;
    const int N = in_sizes[1] / B;                 // 262144
    const float2* inp2 = (const float2*)d_in[0];   // [B, N] of (pred0, pred1)
    const int*    lab  = (const int*)d_in[1];      // [B, N]
    float*        out  = (float*)d_out;

    // workspace layout
    char* ws = (char*)d_ws;
    const size_t off_st    = 0;
    const size_t off_h0    = off_st   + (size_t)B * sizeof(SState);
    const size_t off_h1    = off_h0   + (size_t)B * L0BINS * 4;
    const size_t off_h2    = off_h1   + (size_t)B * L1BINS * 4;
    const size_t off_cbuf  = off_h2   + (size_t)B * L2BINS * 4;
    const size_t off_keys  = off_cbuf + (size_t)B * CAPK * 4;
    const size_t need_pack = off_keys + (size_t)B * N * 4;
    const int use_cbuf = (ws_size >= off_keys) ? 1 : 0;
    const int pack     = (ws_size >= need_pack) ? 1 : 0;

    SState*   st    = (SState*)(ws + off_st);
    unsigned* hist0 = (unsigned*)(ws + off_h0);
    unsigned* hist1 = (unsigned*)(ws + off_h1);
    unsigned* hist2 = (unsigned*)(ws + off_h2);
    unsigned* cbuf  = (unsigned*)(ws + off_cbuf);
    unsigned* keys  = (unsigned*)(ws + off_keys);

    // zero state + histograms (accumulators must reset every call / graph replay)
    {
        size_t words = off_cbuf / 4;
        int blocks = (int)((words + BSZ - 1) / BSZ);
        if (blocks > 4096) blocks = 4096;
        k_init<<<blocks, BSZ, 0, stream>>>((unsigned*)ws, words);
    }

    dim3 gpass(B * CHUNKS);
    k_pass0  <<<gpass, BSZ, 0, stream>>>(inp2, lab, keys, st, hist0, N, pack);
    k_select0<<<B,     BSZ, 0, stream>>>(st, hist0);
    k_pass1  <<<gpass, BSZ, 0, stream>>>(inp2, lab, keys, st, hist1, cbuf, N, pack, use_cbuf);
    k_tail   <<<B,     BSZ, 0, stream>>>(cbuf, st, use_cbuf);
    k_select1<<<B,     BSZ, 0, stream>>>(st, hist1);
    k_pass2  <<<gpass, BSZ, 0, stream>>>(inp2, lab, keys, st, hist2, N, pack);
    k_select2<<<B,     BSZ, 0, stream>>>(st, hist2);
    k_final  <<<1, 64, 0, stream>>>(st, out, B);
}